// Evaluate_24352464569116
// MI455X (gfx1250) — compile-verified
//
#include <hip/hip_runtime.h>

// Problem constants (from reference)
#define B_ 4
#define C_ 64
#define H_ 96
#define W_ 192
#define S_ 12
#define F_ 3
#define I_ 4
#define TEMP 7.0f
#define SCALE (TEMP / (float)C_)   // mean over channels * temperature

#define LPAD 68                    // padded channel stride for transposed left row (bank-conflict-free)
#define NTHREADS 256

typedef __attribute__((ext_vector_type(2))) float v2f;
typedef __attribute__((ext_vector_type(8))) float v8f;

// ---- CDNA5 async global->LDS copies (ASYNCcnt-tracked, bypass VGPRs) ----
__device__ __forceinline__ void async_g2l_b128(const void* gptr, void* lptr) {
    unsigned int laddr = (unsigned int)(unsigned long long)lptr;  // flat->LDS: low 32 bits
    asm volatile("global_load_async_to_lds_b128 %0, %1, off"
                 :: "v"(laddr), "v"(gptr) : "memory");
}
__device__ __forceinline__ void async_g2l_b32(const void* gptr, void* lptr) {
    unsigned int laddr = (unsigned int)(unsigned long long)lptr;
    asm volatile("global_load_async_to_lds_b32 %0, %1, off"
                 :: "v"(laddr), "v"(gptr) : "memory");
}
__device__ __forceinline__ void wait_async_zero() {
    asm volatile("s_wait_asynccnt 0" ::: "memory");
}

__global__ __launch_bounds__(NTHREADS) void stereo_row_kernel(
    const float* __restrict__ left,
    const float* __restrict__ right,
    const float* __restrict__ disp,
    const float* __restrict__ ndisp,
    float* __restrict__ out)
{
    // LDS staging for one (b,h) row
    __shared__ float4 s_right4[(C_ * W_) / 4];   // right row, [c][w] row-major, 48 KB
    __shared__ float  s_leftT[W_ * LPAD];        // left row transposed [w][c], padded, 51 KB
    __shared__ int    s_idx[W_ * S_];            // gather index per (w, s), 9 KB
    __shared__ float  s_str[W_ * S_];            // strengths per (w, s), 9 KB
    float* s_right = (float*)s_right4;

    const int tid = threadIdx.x;
    const int row = blockIdx.x;          // 0 .. B_*H_-1
    const int b = row / H_;
    const int h = row % H_;

    const size_t chanStride = (size_t)H_ * W_;
    const size_t rowBase = ((size_t)(b * C_) * H_ + h) * W_;   // [b][0][h][0] for left/right
    const size_t dBase   = ((size_t)(b * S_) * H_ + h) * W_;   // [b][0][h][0] for disp/ndisp

    // ---- Async-stage right row into LDS ([c][w] row-major), 16B chunks ----
    {
        const int nv = (C_ * W_) / 4;            // 3072 float4, 12 iters/thread, no divergence
        for (int j = tid; j < nv; j += NTHREADS) {
            int c  = j / (W_ / 4);
            int w4 = j % (W_ / 4);
            async_g2l_b128(right + rowBase + (size_t)c * chanStride + (size_t)w4 * 4,
                           &s_right4[j]);
        }
    }
    // ---- Async-stage left row TRANSPOSED [w][c]: coalesced global reads, LDS scatter ----
    for (int j = tid; j < C_ * W_; j += NTHREADS) {   // 48 iters/thread, no divergence
        int c = j / W_, w = j % W_;
        async_g2l_b32(left + rowBase + (size_t)c * chanStride + w,
                      &s_leftT[w * LPAD + c]);
    }
    // ---- Gather indices: idx = (int)clamp(w - d, 0, W-1)  (truncation == floor for >=0) ----
    for (int j = tid; j < S_ * W_; j += NTHREADS) {
        int s = j / W_, w = j % W_;
        float d  = disp[dBase + (size_t)s * chanStride + w];
        float ry = fminf(fmaxf((float)w - d, 0.0f), (float)(W_ - 1));
        s_idx[w * S_ + s] = (int)ry;
    }
    wait_async_zero();     // my async LDS writes are done
    __syncthreads();       // everyone's are

    // ---- Phase 1: strengths via f32 WMMA (16 items per wave-tile, diagonal extraction) ----
    const int lane  = tid & 31;
    const int wave  = tid >> 5;          // wave32
    const int m     = lane & 15;         // A row == B col == item index within tile
    const int kbase = (lane >> 4) << 1;  // lanes 0-15 -> K 0,1 ; lanes 16-31 -> K 2,3
    const int NT    = (W_ * S_) / 16;    // 144 tiles per row

    for (int T = wave; T < NT; T += (NTHREADS / 32)) {
        const int t = T * 16 + m;        // item = pixel*S_ + sample
        const int p = t / S_;            // pixel of this item
        const int g = s_idx[t];          // gathered right column
        const float* aptr  = &s_leftT[p * LPAD + kbase];
        const float* bptr  = &s_right[kbase * W_ + g];

        v8f acc = {};
        #pragma unroll
        for (int kc = 0; kc < C_ / 4; ++kc) {
            v2f a = *(const v2f*)(aptr + kc * 4);        // 8B-aligned (kbase even, LPAD even)
            v2f bb;
            bb.x = bptr[kc * 4 * W_];
            bb.y = bptr[kc * 4 * W_ + W_];
            acc = __builtin_amdgcn_wmma_f32_16x16x4_f32(
                /*neg_a=*/false, a, /*neg_b=*/false, bb,
                /*c_mod=*/(short)0, acc, /*reuse_a=*/false, /*reuse_b=*/false);
        }

        // Diagonal (i,i): i<8 -> VGPR i, lane i ; i>=8 -> VGPR i-8, lane i+16
        int k = (lane < 8) ? lane : (lane - 24);
        float v = acc[0];
        #pragma unroll
        for (int j = 1; j < 8; ++j) v = (k == j) ? acc[j] : v;
        if (lane < 8)        s_str[T * 16 + lane]      = v * SCALE;
        else if (lane >= 24) s_str[T * 16 + lane - 16] = v * SCALE;
    }
    __syncthreads();

    // ---- Phase 2: 3-wide softmax over filter samples, weighted sums, write outputs ----
    const size_t oBase = ((size_t)(b * I_) * H_ + h) * W_;   // into [B, I, H, W]
    const int HALF = B_ * I_ * H_ * W_;                      // ndisp_out first, disp_out second
    for (int j = tid; j < I_ * W_; j += NTHREADS) {
        int i = j / W_, w = j % W_;
        int s0 = i * F_;
        float st0 = s_str[w * S_ + s0 + 0];
        float st1 = s_str[w * S_ + s0 + 1];
        float st2 = s_str[w * S_ + s0 + 2];
        float mx = fmaxf(st0, fmaxf(st1, st2));
        float e0 = __expf(st0 - mx), e1 = __expf(st1 - mx), e2 = __expf(st2 - mx);
        float inv = 1.0f / (e0 + e1 + e2);

        size_t g0 = dBase + (size_t)s0 * chanStride + w;
        float d0 = disp[g0], d1 = disp[g0 + chanStride], d2 = disp[g0 + 2 * chanStride];
        float n0 = ndisp[g0], n1 = ndisp[g0 + chanStride], n2 = ndisp[g0 + 2 * chanStride];

        float dsum = (d0 * e0 + d1 * e1 + d2 * e2) * inv;
        float nsum = (n0 * e0 + n1 * e1 + n2 * e2) * inv;

        size_t oi = oBase + (size_t)i * chanStride + w;
        out[oi]        = nsum;   // ndisp_out (first tuple element)
        out[HALF + oi] = dsum;   // disp_out  (second tuple element)
    }
}

extern "C" void kernel_launch(void* const* d_in, const int* in_sizes, int n_in,
                              void* d_out, int out_size, void* d_ws, size_t ws_size,
                              hipStream_t stream) {
    (void)in_sizes; (void)n_in; (void)out_size; (void)d_ws; (void)ws_size;
    const float* left  = (const float*)d_in[0];
    const float* right = (const float*)d_in[1];
    const float* disp  = (const float*)d_in[2];
    const float* ndisp = (const float*)d_in[3];
    float* out = (float*)d_out;

    dim3 grid(B_ * H_);
    dim3 block(NTHREADS);
    stereo_row_kernel<<<grid, block, 0, stream>>>(left, right, disp, ndisp, out);
}